// NeighborAggregation_23106924053258
// MI455X (gfx1250) — compile-verified
//
#include <hip/hip_runtime.h>
#include <hip/hip_bf16.h>

// NeighborAggregation: out = concat(x, y) with
//   y[b,k,:] = ( x[(i+1)%32,(j+1)%32] + x[(i+1)%32,(j-2)%32]
//              + x[(i-2)%32,(j+1)%32] + x[(i-2)%32,(j-2)%32] - 4*x[i,j] ) * 8^(k-1024)
// 8^(k-1024) underflows to exactly 0.0f for k <= 974 -> y row is exactly 0.
// b=64, k=1024 (32x32), c=512, fp32. Memory-bound: ~384 MB HBM traffic.

typedef float float4v __attribute__((ext_vector_type(4)));

namespace {
constexpr int C  = 512;    // channels
constexpr int KK = 1024;   // k = 32*32
}

__global__ __launch_bounds__(128) void neighbor_agg_kernel(
    const float* __restrict__ x, float* __restrict__ out) {
  const int row = blockIdx.x;          // row = b*1024 + k
  const int k   = row & (KK - 1);
  const int b   = row >> 10;
  const int tid = threadIdx.x;         // 0..127, one float4 (16B) each

  const float* __restrict__ xrow = x + (size_t)row * C;
  float* __restrict__ orow       = out + (size_t)row * (2 * C);

  // Exact power-of-two factor 8^(k-1024); underflows to 0 exactly like the ref.
  const float factor = __builtin_amdgcn_ldexpf(1.0f, 3 * (k - KK));

  if (factor == 0.0f) {
    // ---- 975/1024 of rows: y is exactly zero; only copy x -> out[:,:,0:512].
    // CDNA5 async DMA path: global -> LDS -> global, no VGPR data movement.
    __shared__ float lds[C];
    // Low 32 bits of a flat pointer to LDS == workgroup-relative LDS byte
    // offset (flat LDS aperture truncates to addr[31:0]), which is exactly
    // the addressing the async ops use (the workgroup's LDS base plus the VGPR address plus the instruction offset).
    unsigned lds_addr = (unsigned)(size_t)&lds[tid * 4];
    unsigned goff     = (unsigned)(tid * 16);
    unsigned long long src = (unsigned long long)xrow;
    unsigned long long dst = (unsigned long long)orow;

    asm volatile("global_load_async_to_lds_b128 %0, %1, %2 offset:0"
                 :: "v"(lds_addr), "v"(goff), "s"(src)
                 : "memory");
    asm volatile("s_wait_asynccnt 0x0" ::: "memory");
    // NT store: don't let the 256MB output stream evict x from the 192MB L2.
    asm volatile("global_store_async_from_lds_b128 %0, %1, %2 offset:0 th:TH_STORE_NT"
                 :: "v"(goff), "v"(lds_addr), "s"(dst)
                 : "memory");
    // y half = exactly 0 (finite s * 0.0f factor).
    float4v z = {0.f, 0.f, 0.f, 0.f};
    __builtin_nontemporal_store(z, (float4v*)(orow + C) + tid);
    // Outstanding async ops drain at S_ENDPGM (implicit wait-idle).
  } else {
    // ---- 49/1024 of rows: real stencil. Neighbor rows are L2-resident.
    const int i  = k >> 5, j = k & 31;
    const int ip = (i + 1) & 31, im = (i + 30) & 31;   // i+1, i-2 mod 32
    const int jp = (j + 1) & 31, jm = (j + 30) & 31;   // j+1, j-2 mod 32
    const size_t base = (size_t)b * ((size_t)KK * C);

    const float4v* r0 = (const float4v*)(x + base + (size_t)(ip * 32 + jp) * C);
    const float4v* r1 = (const float4v*)(x + base + (size_t)(ip * 32 + jm) * C);
    const float4v* r2 = (const float4v*)(x + base + (size_t)(im * 32 + jp) * C);
    const float4v* r3 = (const float4v*)(x + base + (size_t)(im * 32 + jm) * C);
    const float4v* rc = (const float4v*)xrow;

    float4v cen = rc[tid];
    float4v s   = r0[tid] + r1[tid] + r2[tid] + r3[tid] - 4.0f * cen;

    __builtin_nontemporal_store(cen,        (float4v*)orow + tid);       // copy half
    __builtin_nontemporal_store(s * factor, (float4v*)(orow + C) + tid); // y half
  }
}

extern "C" void kernel_launch(void* const* d_in, const int* in_sizes, int n_in,
                              void* d_out, int out_size, void* d_ws, size_t ws_size,
                              hipStream_t stream) {
  const float* x = (const float*)d_in[0];
  float* out     = (float*)d_out;

  const int rows = in_sizes[0] / C;    // b * 1024 = 65536 rows
  dim3 grid((unsigned)rows);
  dim3 block(128);
  neighbor_agg_kernel<<<grid, block, 0, stream>>>(x, out);
}